// MVNProfile_78314433675744
// MI455X (gfx1250) — compile-verified
//
#include <hip/hip_runtime.h>
#include <math.h>

// ---------------------------------------------------------------------------
// MVN profile: B=512 rows, D=256 features, N=65536 pixels (16x64x64 grid).
//   means/s    : WMMA f32 16x16x4 GEMM (512x256 @ 256x16, 9 cols used)
//   params     : per-row {mean, inv-diag Cholesky, off-diags} (softmax
//                cancels -1.5*LOG2PI - log_det, so only maha is needed)
//   pass A     : online softmax (running max/sum) per row -> no 128MB spill
//   pass B     : single 128MB nontemporal write of normalized probs
// ---------------------------------------------------------------------------

typedef __attribute__((ext_vector_type(2))) float v2f;
typedef __attribute__((ext_vector_type(4))) float v4f;
typedef __attribute__((ext_vector_type(8))) float v8f;

#define B_ROWS   512
#define D_FEAT   256
#define N_PIX    65536
#define P_STRIDE 12   // padded per-row param stride (9 used)

__device__ __forceinline__ float softplus_f(float x) {
    return (x > 15.0f) ? x : log1pf(__expf(x));
}
__device__ __forceinline__ float elu1_f(float x) {
    // elu(x) + 1  ==  x+1 (x>0)  |  exp(x) (x<=0)
    return (x > 0.0f) ? (x + 1.0f) : __expf(x);
}

// ---------------------------------------------------------------------------
// Kernel 1: one wave (32 lanes) per 16-row tile of the batch.
// WMMA f32 16x16x4, K-loop over D=256 (64 WMMAs). B-matrix = [mean_w;scale_w]^T
// padded to 16 cols. Pad lanes (cols 9..15) load a *valid* row and are zeroed
// by a 0/1 mask multiply -> no divergent branches / exec-mask traffic in the
// inner loop (all 32 lanes issue plain global_load_b64s).
// C tile staged through LDS, then lanes 0..15 each finish one batch row
// (bias + elu + softplus + reciprocals).
// ---------------------------------------------------------------------------
__global__ __launch_bounds__(32)
void mvn_params_kernel(const float* __restrict__ rep,
                       const float* __restrict__ mean_w,
                       const float* __restrict__ mean_b,
                       const float* __restrict__ scale_w,
                       const float* __restrict__ scale_b,
                       float* __restrict__ params) {
    __shared__ float sh[16 * 16];

    const int lane = threadIdx.x;      // 0..31 (wave32)
    const int col  = lane & 15;        // M for A, N for B, col for C
    const int half = lane >> 4;        // K split: 0 -> K{0,1}, 1 -> K{2,3}
    const int m0   = blockIdx.x * 16;

    const float* arow = rep + (size_t)(m0 + col) * D_FEAT;
    // B[k][n] = W[n][k]; n in [0,3) -> mean_w, [3,9) -> scale_w, else pad.
    const int   wr   = (col < 9) ? col : 8;            // clamp to a valid row
    const float mask = (col < 9) ? 1.0f : 0.0f;        // zero the pad columns
    const float* wrow = (wr < 3) ? (mean_w + (size_t)wr * D_FEAT)
                                 : (scale_w + (size_t)(wr - 3) * D_FEAT);

    v8f acc = {};
    for (int k = 0; k < D_FEAT; k += 4) {
        const int kk = k + half * 2;
        v2f a; a.x = arow[kk];        a.y = arow[kk + 1];        // A 16x4
        v2f b; b.x = wrow[kk] * mask; b.y = wrow[kk + 1] * mask; // B 4x16
        acc = __builtin_amdgcn_wmma_f32_16x16x4_f32(
            /*neg_a=*/false, a, /*neg_b=*/false, b,
            /*c_mod=*/(short)0, acc, /*reuse_a=*/false, /*reuse_b=*/false);
    }

    // C layout: acc[j] = C[j + 8*half][col]
#pragma unroll
    for (int j = 0; j < 8; ++j)
        sh[(j + 8 * half) * 16 + col] = acc[j];
    __syncthreads();

    if (lane < 16) {
        const int b = m0 + lane;
        const float* row = sh + lane * 16;
        const float mx = row[0] + mean_b[0];
        const float my = row[1] + mean_b[1];
        const float mz = row[2] + mean_b[2];
        const float l00 = softplus_f(elu1_f(row[3] + scale_b[0]));
        const float l10 =            elu1_f(row[4] + scale_b[1]);
        const float l11 = softplus_f(elu1_f(row[5] + scale_b[2]));
        const float l20 =            elu1_f(row[6] + scale_b[3]);
        const float l21 =            elu1_f(row[7] + scale_b[4]);
        const float l22 = softplus_f(elu1_f(row[8] + scale_b[5]));
        float* p = params + (size_t)b * P_STRIDE;
        p[0] = mx;          p[1] = my;   p[2] = mz;
        p[3] = 1.0f / l00;  p[4] = l10;  p[5] = 1.0f / l11;
        p[6] = l20;         p[7] = l21;  p[8] = 1.0f / l22;
    }
}

__device__ __forceinline__ float maha_f(const float* __restrict__ q,
                                        float mx, float my, float mz,
                                        float i00, float l10, float i11,
                                        float l20, float l21, float i22) {
    const float dx = q[0] - mx, dy = q[1] - my, dz = q[2] - mz;
    const float z0 = dx * i00;
    const float z1 = (dy - l10 * z0) * i11;
    const float z2 = (dz - l20 * z0 - l21 * z1) * i22;
    return fmaf(z0, z0, fmaf(z1, z1, z2 * z2));
}

// ---------------------------------------------------------------------------
// Kernel 2: one block per batch row; online softmax reduction over 65536
// pixels (positions stay resident in L2: 768KB). No 128MB intermediate.
// ---------------------------------------------------------------------------
__global__ __launch_bounds__(256)
void mvn_reduce_kernel(const float* __restrict__ pos,
                       const float* __restrict__ params,
                       float* __restrict__ stats) {
    __shared__ float sm[256];
    __shared__ float ss[256];
    const int b   = blockIdx.x;
    const int tid = threadIdx.x;
    const float* p = params + (size_t)b * P_STRIDE;
    const float mx = p[0], my = p[1], mz = p[2];
    const float i00 = p[3], l10 = p[4], i11 = p[5];
    const float l20 = p[6], l21 = p[7], i22 = p[8];

    float mmax = -3.0e38f, s = 0.0f;
    for (int i = 0; i < N_PIX / 256; ++i) {
        const int n = tid + (i << 8);
        const float lp = -0.5f * maha_f(pos + 3 * n, mx, my, mz,
                                        i00, l10, i11, l20, l21, i22);
        if (lp > mmax) {                    // rare after warmup
            s = s * __expf(mmax - lp) + 1.0f;
            mmax = lp;
        } else {
            s += __expf(lp - mmax);
        }
    }
    sm[tid] = mmax; ss[tid] = s;
    __syncthreads();
    for (int off = 128; off > 0; off >>= 1) {
        if (tid < off) {
            const float m2 = sm[tid + off], s2 = ss[tid + off];
            const float M  = fmaxf(sm[tid], m2);
            ss[tid] = ss[tid] * __expf(sm[tid] - M) + s2 * __expf(m2 - M);
            sm[tid] = M;
        }
        __syncthreads();
    }
    if (tid == 0) {
        stats[b * 2 + 0] = sm[0];
        stats[b * 2 + 1] = 1.0f / (ss[0] + 1e-10f);
    }
}

// ---------------------------------------------------------------------------
// Kernel 3: bandwidth-bound pass. 4 pixels/thread, fully-normalized
// probabilities, coalesced 128-bit *nontemporal* stores (written once,
// never read back -> don't pollute L2).
// ---------------------------------------------------------------------------
__global__ __launch_bounds__(256)
void mvn_write_kernel(const float* __restrict__ pos,
                      const float* __restrict__ params,
                      const float* __restrict__ stats,
                      float* __restrict__ out) {
    const int b = blockIdx.y;
    const float* p = params + (size_t)b * P_STRIDE;   // wave-uniform -> s_load
    const float mx = p[0], my = p[1], mz = p[2];
    const float i00 = p[3], l10 = p[4], i11 = p[5];
    const float l20 = p[6], l21 = p[7], i22 = p[8];
    const float M   = stats[b * 2 + 0];
    const float inv = stats[b * 2 + 1];

    const int n0 = (blockIdx.x * 256 + threadIdx.x) * 4;
    v4f r;
#pragma unroll
    for (int j = 0; j < 4; ++j) {
        const float maha = maha_f(pos + 3 * (n0 + j), mx, my, mz,
                                  i00, l10, i11, l20, l21, i22);
        r[j] = inv * __expf(fmaf(-0.5f, maha, -M));
    }
    __builtin_nontemporal_store(r, (v4f*)(out + (size_t)b * N_PIX + n0));
}

// ---------------------------------------------------------------------------
extern "C" void kernel_launch(void* const* d_in, const int* in_sizes, int n_in,
                              void* d_out, int out_size, void* d_ws, size_t ws_size,
                              hipStream_t stream) {
    const float* rep     = (const float*)d_in[0];   // (512, 256)
    const float* mean_w  = (const float*)d_in[1];   // (3, 256)
    const float* mean_b  = (const float*)d_in[2];   // (3,)
    const float* scale_w = (const float*)d_in[3];   // (6, 256)
    const float* scale_b = (const float*)d_in[4];   // (6,)
    const float* pos     = (const float*)d_in[5];   // (65536, 3)
    float* out = (float*)d_out;                     // (512, 65536)

    float* params = (float*)d_ws;                   // 512 * 12 floats
    float* stats  = params + B_ROWS * P_STRIDE;     // 512 * 2 floats

    mvn_params_kernel<<<B_ROWS / 16, 32, 0, stream>>>(
        rep, mean_w, mean_b, scale_w, scale_b, params);

    mvn_reduce_kernel<<<B_ROWS, 256, 0, stream>>>(pos, params, stats);

    dim3 g3(N_PIX / (256 * 4), B_ROWS);
    mvn_write_kernel<<<g3, 256, 0, stream>>>(pos, params, stats, out);
}